// init_cir_q_28793460753156
// MI455X (gfx1250) — compile-verified
//
#include <hip/hip_runtime.h>
#include <math.h>

// ---------------------------------------------------------------------------
// Z = U X U^H, U = kron of 11 single-qubit gates (2048x2048 complex64).
// U = Ga^(bits10..7) * Gb^(bits6..3) * Gc16^(bits3..0), each group gate 16x16.
// Phase L (one kernel): per 16-column slab, apply all 3 group gates to rows
//   entirely inside LDS (2048x16 complex slab, 272 KB of the 320 KB WGP LDS).
// Phase R (one kernel): per 16-row slab, store transposed in LDS and apply
//   the 3 conjugated group gates (right-mul by G^H == left-mul by conj(G) in
//   transposed space), final sweep writes interleaved complex64 to d_out.
// All 16x16 products run on V_WMMA_F32_16X16X4_F32 (K=16 -> 4 issues/term).
// ---------------------------------------------------------------------------

typedef __attribute__((ext_vector_type(2))) float v2f;
typedef __attribute__((ext_vector_type(8))) float v8f;

#define DIM    2048
#define RS     17                       // padded LDS row stride (floats)
#define LDS_FLOATS (DIM * RS)           // one plane
#define LDS_BYTES  (LDS_FLOATS * 2 * (int)sizeof(float))
#define W_MUL  0.6324555320336759f      // sqrt(2) * 5^-0.5

__device__ inline v8f wmma4(v2f a, v2f b, v8f c) {
  // D(16x16,f32) = A(16x4,f32) x B(4x16,f32) + C
  return __builtin_amdgcn_wmma_f32_16x16x4_f32(
      false, a, false, b, (short)0, c, false, false);
}

// ---------------------------------------------------------------------------
// Build the three 16x16 complex group gates from weight[33].
// Qubit q lives at bit (10-q).  grp0: bits 10..7 (qubits 0..3, stride 128)
// grp1: bits 6..3 (qubits 4..7, stride 8)
// grp2: bits 3..0 (identity on bit 3, qubits 8..10 on bits 2..0, stride 1)
// ---------------------------------------------------------------------------
__global__ void build_gates_kernel(const float* __restrict__ weight,
                                   float2* __restrict__ gates) {
  int t  = threadIdx.x;      // 0..255
  int r  = t >> 4;
  int rp = t & 15;
  for (int grp = 0; grp < 3; ++grp) {
    float accr = 1.0f, acci = 0.0f;
    for (int p = 3; p >= 0; --p) {
      int br = (r  >> p) & 1;
      int bc = (rp >> p) & 1;
      float fr, fi;
      if (grp == 2 && p == 3) {
        fr = (br == bc) ? 1.0f : 0.0f;
        fi = 0.0f;
      } else {
        int q = (grp == 2) ? (8 + (2 - p)) : (grp * 4 + (3 - p));
        float hx = 0.5f * W_MUL * weight[q * 3 + 0];
        float hy = 0.5f * W_MUL * weight[q * 3 + 1];
        float hz = 0.5f * W_MUL * weight[q * 3 + 2];
        float cx = cosf(hx), sx = sinf(hx);
        float cy = cosf(hy), sy = sinf(hy);
        float cz = cosf(hz), sz = sinf(hz);
        float mr, mi;                     // m = Ry @ Rx
        if      (br == 0 && bc == 0) { mr =  cy * cx; mi =  sy * sx; }
        else if (br == 0 && bc == 1) { mr = -sy * cx; mi = -cy * sx; }
        else if (br == 1 && bc == 0) { mr =  sy * cx; mi = -cy * sx; }
        else                         { mr =  cy * cx; mi = -sy * sx; }
        float pr = cz, pi = (br == 0) ? -sz : sz;  // Rz row phase
        fr = pr * mr - pi * mi;
        fi = pr * mi + pi * mr;
      }
      float nr = accr * fr - acci * fi;
      float ni = accr * fi + acci * fr;
      accr = nr; acci = ni;
    }
    gates[grp * 256 + r * 16 + rp] = make_float2(accr, acci);
  }
}

// Gate fragments: A[m=lo][k = ks*4 + j + 2*hi] of the 16x16 group gate,
// pre-negated imag copy supplies the complex-product minus term.
template <bool CONJ>
__device__ inline void load_gate_frags(const float2* __restrict__ g,
                                       int lo, int hi,
                                       v2f (&agr)[4], v2f (&agi)[4], v2f (&agin)[4]) {
#pragma unroll
  for (int ks = 0; ks < 4; ++ks) {
#pragma unroll
    for (int j = 0; j < 2; ++j) {
      int k = ks * 4 + j + 2 * hi;
      float2 gv = g[lo * 16 + k];
      float gi = CONJ ? -gv.y : gv.y;
      agr[ks][j]  = gv.x;
      agi[ks][j]  = gi;
      agin[ks][j] = -gi;
    }
  }
}

// One in-place LDS sweep: out[map(r),c] = sum_k G[r,k] * in[map(k),c],
// map(k) = base + k*2^LGS.  Wave w handles batches w*NB..w*NB+NB-1; each
// batch's read-set equals its write-set, batches partition the 2048 rows.
template <int LGS, bool CONJ, int NB>
__device__ inline void lds_sweep(float* __restrict__ Lr, float* __restrict__ Li,
                                 const float2* __restrict__ gtab,
                                 int w, int lo, int hi) {
  constexpr int S = 1 << LGS;
  v2f agr[4], agi[4], agin[4];
  load_gate_frags<CONJ>(gtab, lo, hi, agr, agi, agin);
#pragma unroll 1
  for (int bb = 0; bb < NB; ++bb) {
    int b    = w * NB + bb;
    int base = ((b >> LGS) << (LGS + 4)) | (b & (S - 1));
    v2f br[4], bi[4];
#pragma unroll
    for (int ks = 0; ks < 4; ++ks)
#pragma unroll
      for (int j = 0; j < 2; ++j) {
        int row = base + (ks * 4 + j + 2 * hi) * S;
        br[ks][j] = Lr[row * RS + lo];
        bi[ks][j] = Li[row * RS + lo];
      }
    v8f dr = {}; v8f di = {};
#pragma unroll
    for (int ks = 0; ks < 4; ++ks) {
      dr = wmma4(agr[ks],  br[ks], dr);   // + Gr*Br
      di = wmma4(agi[ks],  br[ks], di);   // + Gi*Br
      dr = wmma4(agin[ks], bi[ks], dr);   // - Gi*Bi
      di = wmma4(agr[ks],  bi[ks], di);   // + Gr*Bi
    }
#pragma unroll
    for (int j = 0; j < 8; ++j) {
      int row = base + (j + 8 * hi) * S;
      Lr[row * RS + lo] = dr[j];
      Li[row * RS + lo] = di[j];
    }
  }
}

// ---------------------------------------------------------------------------
// Phase L: per 16-column slab, Y = Ga*Gb*Gc16 * X entirely in LDS.
// 512 threads = 16 waves, each wave 8 batches per sweep.
// ---------------------------------------------------------------------------
__global__ __launch_bounds__(512)
void left_fused_kernel(const float* __restrict__ X,
                       float* __restrict__ Yr, float* __restrict__ Yi,
                       const float2* __restrict__ gates) {
  extern __shared__ float lds[];
  float* Lr = lds;
  float* Li = lds + LDS_FLOATS;
  const int lane = threadIdx.x & 31, lo = lane & 15, hi = lane >> 4;
  const int w    = threadIdx.x >> 5;          // 0..15
  const int col0 = blockIdx.x * 16;

  // Sweep 1: grp0 (S=128), real input straight from global, result -> LDS.
  {
    v2f agr[4], agi[4], agin[4];
    load_gate_frags<false>(gates + 0, lo, hi, agr, agi, agin);
#pragma unroll 1
    for (int bb = 0; bb < 8; ++bb) {
      int b = w * 8 + bb;                     // base == b for S=128
      v2f br[4];
#pragma unroll
      for (int ks = 0; ks < 4; ++ks)
#pragma unroll
        for (int j = 0; j < 2; ++j) {
          int row = b + (ks * 4 + j + 2 * hi) * 128;
          br[ks][j] = X[(size_t)row * DIM + col0 + lo];
        }
      v8f dr = {}; v8f di = {};
#pragma unroll
      for (int ks = 0; ks < 4; ++ks) {
        dr = wmma4(agr[ks], br[ks], dr);
        di = wmma4(agi[ks], br[ks], di);
      }
#pragma unroll
      for (int j = 0; j < 8; ++j) {
        int row = b + (j + 8 * hi) * 128;
        Lr[row * RS + lo] = dr[j];
        Li[row * RS + lo] = di[j];
      }
    }
  }
  __syncthreads();
  lds_sweep<3, false, 8>(Lr, Li, gates + 256, w, lo, hi);   // grp1, S=8
  __syncthreads();
  // Sweep 3: grp2 (S=1), LDS -> global planar Y directly from registers.
  {
    v2f agr[4], agi[4], agin[4];
    load_gate_frags<false>(gates + 512, lo, hi, agr, agi, agin);
#pragma unroll 1
    for (int bb = 0; bb < 8; ++bb) {
      int b = w * 8 + bb;
      int base = b * 16;
      v2f br[4], bi[4];
#pragma unroll
      for (int ks = 0; ks < 4; ++ks)
#pragma unroll
        for (int j = 0; j < 2; ++j) {
          int row = base + ks * 4 + j + 2 * hi;
          br[ks][j] = Lr[row * RS + lo];
          bi[ks][j] = Li[row * RS + lo];
        }
      v8f dr = {}; v8f di = {};
#pragma unroll
      for (int ks = 0; ks < 4; ++ks) {
        dr = wmma4(agr[ks],  br[ks], dr);
        di = wmma4(agi[ks],  br[ks], di);
        dr = wmma4(agin[ks], bi[ks], dr);
        di = wmma4(agr[ks],  bi[ks], di);
      }
#pragma unroll
      for (int j = 0; j < 8; ++j) {
        int row = base + j + 8 * hi;
        Yr[(size_t)row * DIM + col0 + lo] = dr[j];
        Yi[(size_t)row * DIM + col0 + lo] = di[j];
      }
    }
  }
}

// ---------------------------------------------------------------------------
// Phase R: per 16-row slab, Z = Y * Ga^H * Gb^H * Gc16^H.  Slab stored
// TRANSPOSED in LDS (L[col][row], padded stride) so the right-mul becomes the
// same left-sweep core with conj(G).  Final sweep writes interleaved
// complex64 to d_out (8 consecutive float2 per lane).
// ---------------------------------------------------------------------------
__global__ __launch_bounds__(512)
void right_fused_kernel(const float* __restrict__ Yr,
                        const float* __restrict__ Yi,
                        float2* __restrict__ outC,
                        const float2* __restrict__ gates) {
  extern __shared__ float lds[];
  float* Lr = lds;
  float* Li = lds + LDS_FLOATS;
  const int lane = threadIdx.x & 31, lo = lane & 15, hi = lane >> 4;
  const int w    = threadIdx.x >> 5;
  const int r0   = blockIdx.x * 16;

  // Preload transposed: L[col][r] = Y[r0+r][col]; coalesced global reads,
  // bank-conflict-free LDS stores (stride RS=17).
#pragma unroll 1
  for (int idx = threadIdx.x; idx < 16 * DIM; idx += 512) {
    int r   = idx >> 11;
    int col = idx & (DIM - 1);
    Lr[col * RS + r] = Yr[(size_t)(r0 + r) * DIM + col];
    Li[col * RS + r] = Yi[(size_t)(r0 + r) * DIM + col];
  }
  __syncthreads();
  lds_sweep<7, true, 8>(Lr, Li, gates + 0,   w, lo, hi);    // Ga^H
  __syncthreads();
  lds_sweep<3, true, 8>(Lr, Li, gates + 256, w, lo, hi);    // Gb^H
  __syncthreads();
  // Final sweep: Gc16^H (S=1), LDS -> interleaved complex64 output.
  {
    v2f agr[4], agi[4], agin[4];
    load_gate_frags<true>(gates + 512, lo, hi, agr, agi, agin);
#pragma unroll 1
    for (int bb = 0; bb < 8; ++bb) {
      int b = w * 8 + bb;
      int base = b * 16;
      v2f br[4], bi[4];
#pragma unroll
      for (int ks = 0; ks < 4; ++ks)
#pragma unroll
        for (int j = 0; j < 2; ++j) {
          int row = base + ks * 4 + j + 2 * hi;
          br[ks][j] = Lr[row * RS + lo];
          bi[ks][j] = Li[row * RS + lo];
        }
      v8f dr = {}; v8f di = {};
#pragma unroll
      for (int ks = 0; ks < 4; ++ks) {
        dr = wmma4(agr[ks],  br[ks], dr);
        di = wmma4(agi[ks],  br[ks], di);
        dr = wmma4(agin[ks], bi[ks], dr);
        di = wmma4(agr[ks],  bi[ks], di);
      }
      // transposed space: D row = Z column, lane lo = Z row (r0+lo)
#pragma unroll
      for (int j = 0; j < 8; ++j) {
        int col = base + j + 8 * hi;
        outC[(size_t)(r0 + lo) * DIM + col] = make_float2(dr[j], di[j]);
      }
    }
  }
}

// ---------------------------------------------------------------------------
extern "C" void kernel_launch(void* const* d_in, const int* in_sizes, int n_in,
                              void* d_out, int out_size, void* d_ws, size_t ws_size,
                              hipStream_t stream) {
  const float* x      = (const float*)d_in[0];   // (2048,2048) f32
  const float* weight = (const float*)d_in[1];   // (33,) f32
  float2* outC        = (float2*)d_out;          // 2048x2048 complex64

  const size_t plane = (size_t)DIM * DIM;
  float* Yr = (float*)d_ws;
  float* Yi = Yr + plane;
  float2* gates = (float2*)(Yi + plane);         // 3 * 256 complex

  build_gates_kernel<<<1, 256, 0, stream>>>(weight, gates);
  left_fused_kernel <<<128, 512, LDS_BYTES, stream>>>(x, Yr, Yi, gates);
  right_fused_kernel<<<128, 512, LDS_BYTES, stream>>>(Yr, Yi, outC, gates);
}